// GTN_72679436583060
// MI455X (gfx1250) — compile-verified
//
#include <hip/hip_runtime.h>

typedef __attribute__((ext_vector_type(2))) float v2f;
typedef __attribute__((ext_vector_type(8))) float v8f;

#define NN 2048
#define BM 128
#define BN 128
#define BK 16
#define THREADS 256
#define APAD 20    // A-tile LDS row stride (floats): 16B-aligned, conflict-free
#define BPAD 136   // B-tile LDS row stride (floats): 16B-aligned, halves 16 banks apart

#if defined(__gfx1250__) && \
    __has_builtin(__builtin_amdgcn_global_load_async_to_lds_b128) && \
    __has_builtin(__builtin_amdgcn_s_wait_asynccnt)
#define USE_ASYNC 1
// param type leaked by the round-2 diagnostic:
//   __attribute__((__vector_size__(4 * sizeof(int)))) int *
typedef int v4i_async __attribute__((vector_size(4 * sizeof(int))));
typedef __attribute__((address_space(3))) v4i_async* lds_v4i_ptr;
#else
#define USE_ASYNC 0
#endif

__device__ __forceinline__ v8f wmma_f32_16x16x4(v2f a, v2f b, v8f c) {
    // 8 args: (neg_a, A, neg_b, B, c_mod, C, reuse_a, reuse_b)
    return __builtin_amdgcn_wmma_f32_16x16x4_f32(
        false, a, false, b, (short)0, c, false, false);
}

// MODE 0: C = 4 * (Am @ Bm)                       (Hpre = (2A)@(2A))
// MODE 1: C = dinv[row] * ((Am @ Bm) + Aadd)      (out  = (2/deg)*(Hpre@A + A))
template <int MODE>
__global__ __launch_bounds__(THREADS)
void gemm_wmma(const float* __restrict__ Am, const float* __restrict__ Bm,
               float* __restrict__ Cm, const float* __restrict__ Aadd,
               const float* __restrict__ dinv)
{
    __shared__ float As[2][BM][APAD];
    __shared__ float Bs[2][BK][BPAD];

    const int tid  = threadIdx.x;
    const int lane = tid & 31;
    const int wave = tid >> 5;
    const int wrow = wave >> 2;        // 0..1  -> 64-row slab
    const int wcol = wave & 3;         // 0..3  -> 32-col slab
    const int mrow = lane & 15;        // A-frag row within tile
    const int half = lane >> 4;        // 0/1: K half (A/B), M half (C/D)
    const int ncol = lane & 15;        // B/C-frag col within tile

    const int blockM = blockIdx.y * BM;
    const int blockN = blockIdx.x * BN;

    v8f acc[4][2];
#pragma unroll
    for (int mi = 0; mi < 4; ++mi)
#pragma unroll
        for (int ni = 0; ni < 2; ++ni)
            acc[mi][ni] = (v8f){};

    // staging indices: each thread moves 2x16B of A-tile and 2x16B of B-tile
    const int a_r = tid >> 2;              // 0..63 (+64)
    const int a_c = (tid & 3) * 4;         // 0,4,8,12
    const int b_r = tid >> 5;              // 0..7  (+8)
    const int b_c = (tid & 31) * 4;        // 0..124

    auto stage = [&](int buf, int kb) {
#if USE_ASYNC
#pragma unroll
        for (int rr = 0; rr < 2; ++rr) {
            const int r = a_r + rr * 64;
            __builtin_amdgcn_global_load_async_to_lds_b128(
                (v4i_async*)&Am[(size_t)(blockM + r) * NN + kb + a_c],
                (lds_v4i_ptr)&As[buf][r][a_c], 0, 0);
        }
#pragma unroll
        for (int rr = 0; rr < 2; ++rr) {
            const int r = b_r + rr * 8;
            __builtin_amdgcn_global_load_async_to_lds_b128(
                (v4i_async*)&Bm[(size_t)(kb + r) * NN + blockN + b_c],
                (lds_v4i_ptr)&Bs[buf][r][b_c], 0, 0);
        }
#else
#pragma unroll
        for (int rr = 0; rr < 2; ++rr) {
            const int r = a_r + rr * 64;
            const float4 v = *reinterpret_cast<const float4*>(
                &Am[(size_t)(blockM + r) * NN + kb + a_c]);
            *reinterpret_cast<float4*>(&As[buf][r][a_c]) = v;
        }
#pragma unroll
        for (int rr = 0; rr < 2; ++rr) {
            const int r = b_r + rr * 8;
            const float4 v = *reinterpret_cast<const float4*>(
                &Bm[(size_t)(kb + r) * NN + blockN + b_c]);
            *reinterpret_cast<float4*>(&Bs[buf][r][b_c]) = v;
        }
#endif
    };

    // prologue: stage chunk 0, wait, barrier
    stage(0, 0);
#if USE_ASYNC
    __builtin_amdgcn_s_wait_asynccnt(0);
#endif
    __syncthreads();

    const int NCHUNK = NN / BK;  // 128
    for (int k = 0; k < NCHUNK; ++k) {
        const int buf = k & 1;
        // prefetch next chunk into the other buffer (freed by previous barrier)
        if (k + 1 < NCHUNK) stage(buf ^ 1, (k + 1) * BK);

#pragma unroll
        for (int kk = 0; kk < BK; kk += 4) {
            v2f afr[4], bfr[2];
#pragma unroll
            for (int mi = 0; mi < 4; ++mi) {
                const float* ap = &As[buf][wrow * 64 + mi * 16 + mrow][kk + half * 2];
                v2f t; t.x = ap[0]; t.y = ap[1];
                afr[mi] = t;
            }
#pragma unroll
            for (int ni = 0; ni < 2; ++ni) {
                v2f t;
                t.x = Bs[buf][kk + half * 2 + 0][wcol * 32 + ni * 16 + ncol];
                t.y = Bs[buf][kk + half * 2 + 1][wcol * 32 + ni * 16 + ncol];
                bfr[ni] = t;
            }
#pragma unroll
            for (int mi = 0; mi < 4; ++mi)
#pragma unroll
                for (int ni = 0; ni < 2; ++ni)
                    acc[mi][ni] = wmma_f32_16x16x4(afr[mi], bfr[ni], acc[mi][ni]);
        }

#if USE_ASYNC
        __builtin_amdgcn_s_wait_asynccnt(0);   // own prefetch done (overlapped compute)
#endif
        __syncthreads();                       // everyone done reading buf, writes visible
    }

    // epilogue: D layout = VGPR r -> row (half*8 + r), lane&15 -> col
#pragma unroll
    for (int mi = 0; mi < 4; ++mi) {
#pragma unroll
        for (int ni = 0; ni < 2; ++ni) {
#pragma unroll
            for (int r = 0; r < 8; ++r) {
                const int row = blockM + wrow * 64 + mi * 16 + half * 8 + r;
                const int col = blockN + wcol * 32 + ni * 16 + ncol;
                const float v = acc[mi][ni][r];
                if (MODE == 0) {
                    Cm[(size_t)row * NN + col] = 4.0f * v;
                } else {
                    Cm[(size_t)row * NN + col] =
                        dinv[row] * (v + Aadd[(size_t)row * NN + col]);
                }
            }
        }
    }
}

// one block per row: deg[i] = sum_j Hpre[i,j] + 1 ; dinv[i] = 2/deg (guarded)
__global__ __launch_bounds__(256)
void rowsum_inv(const float* __restrict__ H, float* __restrict__ dinv)
{
    __shared__ float red[256];
    const int row = blockIdx.x;
    float s = 0.0f;
    for (int c = threadIdx.x * 4; c < NN; c += 256 * 4) {
        const float4 v = *reinterpret_cast<const float4*>(&H[(size_t)row * NN + c]);
        s += (v.x + v.y) + (v.z + v.w);
    }
    red[threadIdx.x] = s;
    __syncthreads();
    for (int off = 128; off > 0; off >>= 1) {
        if (threadIdx.x < off) red[threadIdx.x] += red[threadIdx.x + off];
        __syncthreads();
    }
    if (threadIdx.x == 0) {
        float deg = red[0] + 1.0f;             // + identity
        if (deg <= 1e-10f) deg = 1.0f;
        dinv[row] = 2.0f / deg;                // folds the trailing (2A) factor
    }
}

extern "C" void kernel_launch(void* const* d_in, const int* in_sizes, int n_in,
                              void* d_out, int out_size, void* d_ws, size_t ws_size,
                              hipStream_t stream)
{
    (void)in_sizes; (void)n_in; (void)out_size; (void)ws_size;
    // softmax over a singleton axis == 1.0 exactly, so every GTConv is 2*A and
    // the weight inputs d_in[1..3] are mathematically irrelevant.
    const float* A    = (const float*)d_in[0];
    float*       out  = (float*)d_out;
    float*       Hpre = (float*)d_ws;                                  // 16 MB
    float*       dinv = (float*)((char*)d_ws + (size_t)NN * NN * 4);   // 8 KB

    dim3 grid(NN / BN, NN / BM);
    dim3 block(THREADS);

    gemm_wmma<0><<<grid, block, 0, stream>>>(A, A, Hpre, nullptr, nullptr);
    rowsum_inv<<<NN, 256, 0, stream>>>(Hpre, dinv);
    gemm_wmma<1><<<grid, block, 0, stream>>>(Hpre, A, out, A, dinv);
}